// GIN_small_49160195670360
// MI455X (gfx1250) — compile-verified
//
#include <hip/hip_runtime.h>
#include <hip/hip_bf16.h>

#define NNODES 100000
#define NEDGES 1600000
#define FIN    128
#define HID    32
#define NCLS   10
#define NGRAPH 1000
#define BN_EPS 1e-5f

typedef float v2f __attribute__((ext_vector_type(2)));
typedef float v8f __attribute__((ext_vector_type(8)));

__device__ __forceinline__ v8f wmma_f32_k4(v2f a, v2f b, v8f c) {
  // D(16x16,f32) += A(16x4,f32) * B(4x16,f32); 8-arg VOP3P form
  return __builtin_amdgcn_wmma_f32_16x16x4_f32(
      /*neg_a=*/false, a, /*neg_b=*/false, b,
      /*c_mod=*/(short)0, c, /*reuse_a=*/false, /*reuse_b=*/false);
}

// -------- pre: h = relu(bn(x @ pre_W + pre_b)) , [N,128]@[128,32] --------
__global__ void pre_kernel(const float* __restrict__ x, const float* __restrict__ W,
                           const float* __restrict__ bias,
                           const float* __restrict__ g, const float* __restrict__ bb,
                           const float* __restrict__ mm, const float* __restrict__ vv,
                           float* __restrict__ h) {
  const int wid  = (blockIdx.x * blockDim.x + threadIdx.x) >> 5;
  const int lane = threadIdx.x & 31;
  if (wid >= NNODES / 16) return;           // whole wave exits -> EXEC all-1s for WMMA
  const int m0   = wid * 16;
  const int half = lane >> 4;
  const int l16  = lane & 15;
  const int arow = m0 + l16;
  const float* xrow = x + (size_t)arow * FIN;

  v8f acc0 = {}; v8f acc1 = {};
#pragma unroll 4
  for (int k = 0; k < FIN; k += 4) {
    const int ka = k + half * 2;            // A: lanes 0-15 -> K=k,k+1 ; lanes 16-31 -> K=k+2,k+3
    v2f a;  a.x = xrow[ka];  a.y = xrow[ka + 1];
    v2f b0, b1;                             // B: lane = N col, half selects K pair
    b0.x = W[ka * HID + l16];        b0.y = W[(ka + 1) * HID + l16];
    b1.x = W[ka * HID + 16 + l16];   b1.y = W[(ka + 1) * HID + 16 + l16];
    acc0 = wmma_f32_k4(a, b0, acc0);
    acc1 = wmma_f32_k4(a, b1, acc1);
  }
  const int c0 = l16, c1 = 16 + l16;
  const float s0 = g[c0] * rsqrtf(vv[c0] + BN_EPS);
  const float s1 = g[c1] * rsqrtf(vv[c1] + BN_EPS);
  const float t0 = bb[c0] + (bias[c0] - mm[c0]) * s0;   // fold bias into BN shift
  const float t1 = bb[c1] + (bias[c1] - mm[c1]) * s1;
#pragma unroll
  for (int r = 0; r < 8; ++r) {
    const int row = m0 + r + half * 8;      // D layout: vgpr r -> M=r (lanes<16) / M=r+8
    h[row * HID + c0] = fmaxf(acc0[r] * s0 + t0, 0.f);
    h[row * HID + c1] = fmaxf(acc1[r] * s1 + t1, 0.f);
  }
}

// -------- agg[dst] += h[src] : one wave per edge, lane = feature (128B coalesced) ---
__global__ void scatter_add_kernel(const int* __restrict__ ei, const float* __restrict__ h,
                                   float* __restrict__ agg) {
  const int e    = (blockIdx.x * blockDim.x + threadIdx.x) >> 5;
  const int lane = threadIdx.x & 31;
  if (e >= NEDGES) return;
  const int src = ei[e];
  const int dst = ei[NEDGES + e];
  atomicAdd(agg + (size_t)dst * HID + lane, h[(size_t)src * HID + lane]);
}

__global__ void zero_kernel(float4* __restrict__ p, int n4) {
  const int i = blockIdx.x * blockDim.x + threadIdx.x;
  if (i < n4) p[i] = make_float4(0.f, 0.f, 0.f, 0.f);
}

// -------- fused GIN MLP: z=h+agg; y=mid(z@W1+b1); h=relu(bn(y@W2+b2)) --------
// mid = relu (conv0) or bn+relu (conv1..3). One wave per 16 nodes; LDS transposes
// the D-fragment layout of matmul1 back into A-fragment layout for matmul2.
__global__ void gin_mlp_kernel(const float* h, const float* __restrict__ agg,
                               const float* __restrict__ W1, const float* __restrict__ b1,
                               const float* __restrict__ mg, const float* __restrict__ mb,
                               const float* __restrict__ mmn, const float* __restrict__ mv,
                               int has_mid_bn,
                               const float* __restrict__ W2, const float* __restrict__ b2,
                               const float* __restrict__ og, const float* __restrict__ ob,
                               const float* __restrict__ om, const float* __restrict__ ov,
                               float* hout) {
  __shared__ float lds[16][HID + 1];
  const int lane = threadIdx.x & 31;        // blockDim == 32 (one wave)
  const int m0   = blockIdx.x * 16;
  const int half = lane >> 4;
  const int l16  = lane & 15;
  const int arow = m0 + l16;
  const int c0 = l16, c1 = 16 + l16;

  v8f acc0 = {}; v8f acc1 = {};
#pragma unroll
  for (int s = 0; s < HID / 4; ++s) {
    const int ka = s * 4 + half * 2;
    v2f a;
    a.x = h[arow * HID + ka]     + agg[arow * HID + ka];
    a.y = h[arow * HID + ka + 1] + agg[arow * HID + ka + 1];
    v2f bf0, bf1;
    bf0.x = W1[ka * HID + l16];        bf0.y = W1[(ka + 1) * HID + l16];
    bf1.x = W1[ka * HID + 16 + l16];   bf1.y = W1[(ka + 1) * HID + 16 + l16];
    acc0 = wmma_f32_k4(a, bf0, acc0);
    acc1 = wmma_f32_k4(a, bf1, acc1);
  }
  float s0 = 1.f, s1 = 1.f, t0 = b1[c0], t1 = b1[c1];
  if (has_mid_bn) {
    s0 = mg[c0] * rsqrtf(mv[c0] + BN_EPS);
    s1 = mg[c1] * rsqrtf(mv[c1] + BN_EPS);
    t0 = mb[c0] + (b1[c0] - mmn[c0]) * s0;
    t1 = mb[c1] + (b1[c1] - mmn[c1]) * s1;
  }
#pragma unroll
  for (int r = 0; r < 8; ++r) {
    const int row = r + half * 8;
    lds[row][c0] = fmaxf(acc0[r] * s0 + t0, 0.f);
    lds[row][c1] = fmaxf(acc1[r] * s1 + t1, 0.f);
  }
  __syncthreads();

  v8f d0 = {}; v8f d1 = {};
#pragma unroll
  for (int s = 0; s < HID / 4; ++s) {
    const int ka = s * 4 + half * 2;
    v2f a;  a.x = lds[l16][ka];  a.y = lds[l16][ka + 1];
    v2f bf0, bf1;
    bf0.x = W2[ka * HID + l16];        bf0.y = W2[(ka + 1) * HID + l16];
    bf1.x = W2[ka * HID + 16 + l16];   bf1.y = W2[(ka + 1) * HID + 16 + l16];
    d0 = wmma_f32_k4(a, bf0, d0);
    d1 = wmma_f32_k4(a, bf1, d1);
  }
  const float os0 = og[c0] * rsqrtf(ov[c0] + BN_EPS);
  const float os1 = og[c1] * rsqrtf(ov[c1] + BN_EPS);
  const float ot0 = ob[c0] + (b2[c0] - om[c0]) * os0;
  const float ot1 = ob[c1] + (b2[c1] - om[c1]) * os1;
#pragma unroll
  for (int r = 0; r < 8; ++r) {
    const int row = m0 + r + half * 8;
    hout[row * HID + c0] = fmaxf(d0[r] * os0 + ot0, 0.f);
    hout[row * HID + c1] = fmaxf(d1[r] * os1 + ot1, 0.f);
  }
}

// -------- global_add_pool: wave per node, lane = feature --------
__global__ void pool_kernel(const float* __restrict__ h, const int* __restrict__ batch,
                            float* __restrict__ gpool) {
  const int node = (blockIdx.x * blockDim.x + threadIdx.x) >> 5;
  const int lane = threadIdx.x & 31;
  if (node >= NNODES) return;
  atomicAdd(gpool + (size_t)batch[node] * HID + lane, h[(size_t)node * HID + lane]);
}

// -------- head: relu(gp@post_W+b) -> relu(bn) -> @ro_W + ro_b --------
__global__ void head_kernel(const float* __restrict__ gpool,
                            const float* __restrict__ pW, const float* __restrict__ pb,
                            const float* __restrict__ g,  const float* __restrict__ bb,
                            const float* __restrict__ mm, const float* __restrict__ vv,
                            const float* __restrict__ roW, const float* __restrict__ rob,
                            float* __restrict__ out) {
  __shared__ float lds[16][HID + 1];
  const int lane = threadIdx.x & 31;
  const int m0   = blockIdx.x * 16;
  const int half = lane >> 4;
  const int l16  = lane & 15;
  const int c0 = l16, c1 = 16 + l16;
  const int arow = min(m0 + l16, NGRAPH - 1);   // clamp loads; all lanes stay active

  v8f acc0 = {}; v8f acc1 = {};
#pragma unroll
  for (int s = 0; s < HID / 4; ++s) {
    const int ka = s * 4 + half * 2;
    v2f a;  a.x = gpool[arow * HID + ka];  a.y = gpool[arow * HID + ka + 1];
    v2f bf0, bf1;
    bf0.x = pW[ka * HID + l16];        bf0.y = pW[(ka + 1) * HID + l16];
    bf1.x = pW[ka * HID + 16 + l16];   bf1.y = pW[(ka + 1) * HID + 16 + l16];
    acc0 = wmma_f32_k4(a, bf0, acc0);
    acc1 = wmma_f32_k4(a, bf1, acc1);
  }
  const float s0 = g[c0] * rsqrtf(vv[c0] + BN_EPS);
  const float s1 = g[c1] * rsqrtf(vv[c1] + BN_EPS);
#pragma unroll
  for (int r = 0; r < 8; ++r) {
    const int row = r + half * 8;
    float y0 = fmaxf(acc0[r] + pb[c0], 0.f);
    float y1 = fmaxf(acc1[r] + pb[c1], 0.f);
    lds[row][c0] = fmaxf((y0 - mm[c0]) * s0 + bb[c0], 0.f);
    lds[row][c1] = fmaxf((y1 - mm[c1]) * s1 + bb[c1], 0.f);
  }
  __syncthreads();

  v8f d0 = {};
#pragma unroll
  for (int s = 0; s < HID / 4; ++s) {
    const int ka = s * 4 + half * 2;
    v2f a;  a.x = lds[l16][ka];  a.y = lds[l16][ka + 1];
    v2f bf;                                   // ro_W is [32,10]; pad cols 10..15 with 0
    bf.x = (l16 < NCLS) ? roW[ka * NCLS + l16]       : 0.f;
    bf.y = (l16 < NCLS) ? roW[(ka + 1) * NCLS + l16] : 0.f;
    d0 = wmma_f32_k4(a, bf, d0);
  }
#pragma unroll
  for (int r = 0; r < 8; ++r) {
    const int row = m0 + r + half * 8;
    if (row < NGRAPH && l16 < NCLS)
      out[row * NCLS + l16] = d0[r] + rob[l16];
  }
}

extern "C" void kernel_launch(void* const* d_in, const int* in_sizes, int n_in,
                              void* d_out, int out_size, void* d_ws, size_t ws_size,
                              hipStream_t stream) {
  const float* x        = (const float*)d_in[0];
  const int*   ei       = (const int*)d_in[1];
  const int*   batch    = (const int*)d_in[2];
  const float* pre_W    = (const float*)d_in[3];
  const float* pre_b    = (const float*)d_in[4];
  const float* bnpre_g  = (const float*)d_in[5];
  const float* bnpre_b  = (const float*)d_in[6];
  const float* bnpre_m  = (const float*)d_in[7];
  const float* bnpre_v  = (const float*)d_in[8];
  const float* conv0_W1 = (const float*)d_in[9];
  const float* conv0_b1 = (const float*)d_in[10];
  const float* conv0_W2 = (const float*)d_in[11];
  const float* conv0_b2 = (const float*)d_in[12];
  const float* convk_W1 = (const float*)d_in[13];
  const float* convk_b1 = (const float*)d_in[14];
  const float* ck_bn_g  = (const float*)d_in[15];
  const float* ck_bn_b  = (const float*)d_in[16];
  const float* ck_bn_m  = (const float*)d_in[17];
  const float* ck_bn_v  = (const float*)d_in[18];
  const float* convk_W2 = (const float*)d_in[19];
  const float* convk_b2 = (const float*)d_in[20];
  const float* bnc_g    = (const float*)d_in[21];
  const float* bnc_b    = (const float*)d_in[22];
  const float* bnc_m    = (const float*)d_in[23];
  const float* bnc_v    = (const float*)d_in[24];
  const float* post_W   = (const float*)d_in[25];
  const float* post_b   = (const float*)d_in[26];
  const float* bnpost_g = (const float*)d_in[27];
  const float* bnpost_b = (const float*)d_in[28];
  const float* bnpost_m = (const float*)d_in[29];
  const float* bnpost_v = (const float*)d_in[30];
  const float* ro_W     = (const float*)d_in[31];
  const float* ro_b     = (const float*)d_in[32];
  (void)in_sizes; (void)n_in; (void)out_size; (void)ws_size;
  (void)bnpre_g; // used below

  float* h     = (float*)d_ws;                  // N*32 f32
  float* agg   = h + (size_t)NNODES * HID;      // N*32 f32
  float* gpool = agg + (size_t)NNODES * HID;    // G*32 f32
  float* out   = (float*)d_out;

  // 1) pre-layer GEMM + BN + ReLU
  {
    const int waves = NNODES / 16;              // 6250, exact
    const int thr = 256;
    const int blk = (waves * 32 + thr - 1) / thr;
    pre_kernel<<<blk, thr, 0, stream>>>(x, pre_W, pre_b,
                                        bnpre_g, bnpre_b, bnpre_m, bnpre_v, h);
  }

  // 2) four GIN layers: zero agg -> scatter-add over edges -> fused MLP
  const int n4_agg = NNODES * HID / 4;
  for (int l = 0; l < 4; ++l) {
    zero_kernel<<<(n4_agg + 255) / 256, 256, 0, stream>>>((float4*)agg, n4_agg);
    scatter_add_kernel<<<(NEDGES * 32 + 255) / 256, 256, 0, stream>>>(ei, h, agg);
    if (l == 0) {
      gin_mlp_kernel<<<NNODES / 16, 32, 0, stream>>>(
          h, agg, conv0_W1, conv0_b1,
          bnc_g, bnc_b, bnc_m, bnc_v, /*has_mid_bn=*/0,
          conv0_W2, conv0_b2,
          bnc_g, bnc_b, bnc_m, bnc_v, h);       // out-BN = bnconv[0]
    } else {
      const int k = l - 1;
      gin_mlp_kernel<<<NNODES / 16, 32, 0, stream>>>(
          h, agg,
          convk_W1 + k * HID * HID, convk_b1 + k * HID,
          ck_bn_g + k * HID, ck_bn_b + k * HID, ck_bn_m + k * HID, ck_bn_v + k * HID, 1,
          convk_W2 + k * HID * HID, convk_b2 + k * HID,
          bnc_g + l * HID, bnc_b + l * HID, bnc_m + l * HID, bnc_v + l * HID, h);
    }
  }

  // 3) global add pool + head
  const int n4_gp = NGRAPH * HID / 4;
  zero_kernel<<<(n4_gp + 255) / 256, 256, 0, stream>>>((float4*)gpool, n4_gp);
  pool_kernel<<<(NNODES * 32 + 255) / 256, 256, 0, stream>>>(h, batch, gpool);
  head_kernel<<<(NGRAPH + 15) / 16, 32, 0, stream>>>(
      gpool, post_W, post_b, bnpost_g, bnpost_b, bnpost_m, bnpost_v, ro_W, ro_b, out);
}